// TurboQuantRotation_61838939127831
// MI455X (gfx1250) — compile-verified
//
#include <hip/hip_runtime.h>

// ---------------------------------------------------------------------------
// Fused 2-pass randomized Hadamard transform for MI455X (gfx1250).
//   out = H * (s1 .* (H * (s0 .* x)))      H = orthonormal WHT, D = 4096
// H_4096 = H16 (x) H16 (x) H16  -> per token: three batches of 16x16x16
// matmuls on V_WMMA_F32_16X16X4_F32 (exact fp32: H entries are +-1 and the
// 1/64 orthonormal scale is a power of two).  Memory-bound: 256 MB total ->
// ~11 us at 23.3 TB/s; x is read once, both passes run out of LDS.
// ---------------------------------------------------------------------------

typedef __attribute__((ext_vector_type(2))) float v2f;
typedef __attribute__((ext_vector_type(8))) float v8f;

#define DIM    4096
#define ROWPAD 17            // 16+1: stride-16 LDS access hits 16 distinct banks
#define IBLK   (16 * ROWPAD) // floats per padded 16x16 i-block
#define SCALE  0.015625f     // 4096^-0.5 = 1/64 (exact)

__device__ __forceinline__ float hval(int r, int c) {
    // Sylvester H16 entry: (-1)^popcount(r&c); symmetric in (r,c)
    return (__builtin_popcount(r & c) & 1) ? -1.0f : 1.0f;
}

__global__ void __launch_bounds__(128)
fwht2_kernel(const float* __restrict__ x, const float* __restrict__ signs,
             float* __restrict__ out) {
    __shared__ float lx[16 * IBLK];  // working tile (padded 16x16x16)
    __shared__ float ls[16 * IBLK];  // signs[1], padded identically

    const int token = blockIdx.x;
    const int tid   = threadIdx.x;
    const int lane  = tid & 31;
    const int l15   = lane & 15;
    const int hi    = (lane >> 4) & 1;                     // lane-half selector
    const int wave  = __builtin_amdgcn_readfirstlane(tid >> 5);  // uniform SGPR

    // ---- precompute H16 fragments once; reused by every stage of both passes.
    // hb[c] is simultaneously the B-fragment (K=k', N=l15) for stage K and the
    // A-fragment (M=l15, K=k') for stages J/I (H16 is symmetric).
    v2f hb[4], hbs[4];
#pragma unroll
    for (int c = 0; c < 4; ++c) {
        const int k0 = c * 4 + hi * 2;                     // ISA f32 K split 0/2
        hb[c][0]  = hval(k0,     l15);
        hb[c][1]  = hval(k0 + 1, l15);
        hbs[c][0] = hb[c][0] * SCALE;                      // stage I folds 1/64
        hbs[c][1] = hb[c][1] * SCALE;
    }

    // lane-constant LDS sub-offsets (scalar block base added per block)
    const int aoff = l15 * ROWPAD + hi * 2;   // stage K A: row=l15, K base
    const int boff = hi * 2 * ROWPAD + l15;   // stage J B: row=K, col=l15
    const int coff = hi * 8 * ROWPAD + l15;   // stage K/J D: M=v+8*hi, N=l15
    const int ioff = hi * 2 * IBLK + l15;     // stage I B: row=i', col in j-block
    const int doff = hi * 8 * IBLK + l15;     // stage I D: M=i rows

    const float* __restrict__ xrow = x + (size_t)token * DIM;
    const float* __restrict__ s0   = signs;
    const float* __restrict__ s1   = signs + DIM;

    // ---- load: lx = x .* signs0 ; ls = signs1 (coalesced float4) ----------
#pragma unroll
    for (int it = 0; it < 8; ++it) {
        const int g = (it * 128 + tid) * 4;                // 0..4092, step 4
        const float4 xv = *(const float4*)(xrow + g);
        const float4 av = *(const float4*)(s0 + g);
        const float4 bv = *(const float4*)(s1 + g);
        const int base = (g >> 8) * IBLK + ((g >> 4) & 15) * ROWPAD + (g & 15);
        lx[base + 0] = xv.x * av.x;  lx[base + 1] = xv.y * av.y;
        lx[base + 2] = xv.z * av.z;  lx[base + 3] = xv.w * av.w;
        ls[base + 0] = bv.x;  ls[base + 1] = bv.y;
        ls[base + 2] = bv.z;  ls[base + 3] = bv.w;
    }
    __syncthreads();

#pragma unroll
    for (int pass = 0; pass < 2; ++pass) {
        // ===== stage K: per i-block, X_i <- X_i * H (axis k); wave-local ====
#pragma unroll
        for (int u = 0; u < 4; ++u) {
            const int base = (wave + u * 4) * IBLK;        // scalar block base
            v8f acc = {};
#pragma unroll
            for (int c = 0; c < 4; ++c) {
                const int ab = base + aoff + c * 4;
                v2f a;
                if (pass == 0) {
                    a[0] = lx[ab];          a[1] = lx[ab + 1];
                } else {                    // fold elementwise signs1 into pass 2
                    a[0] = lx[ab] * ls[ab]; a[1] = lx[ab + 1] * ls[ab + 1];
                }
                acc = __builtin_amdgcn_wmma_f32_16x16x4_f32(
                    false, a, false, hb[c], (short)0, acc, false, false);
            }
#pragma unroll
            for (int v = 0; v < 8; ++v)
                lx[base + coff + v * ROWPAD] = acc[v];
        }
        // no barrier: stage J reads exactly the rows this wave just wrote

        // ===== stage J: per i-block, X_i <- H * X_i (axis j); wave-local ====
#pragma unroll
        for (int u = 0; u < 4; ++u) {
            const int base = (wave + u * 4) * IBLK;
            v8f acc = {};
#pragma unroll
            for (int c = 0; c < 4; ++c) {
                v2f b;
                b[0] = lx[base + boff + (c * 4    ) * ROWPAD];
                b[1] = lx[base + boff + (c * 4 + 1) * ROWPAD];
                acc = __builtin_amdgcn_wmma_f32_16x16x4_f32(
                    false, hb[c], false, b, (short)0, acc, false, false);
            }
#pragma unroll
            for (int v = 0; v < 8; ++v)
                lx[base + coff + v * ROWPAD] = acc[v];
        }
        __syncthreads();   // stage I mixes i-blocks across waves

        // ===== stage I: per j-column, X[:,j,:] <- (H/64) * X[:,j,:] =========
#pragma unroll
        for (int u = 0; u < 4; ++u) {
            const int jbase = (wave + u * 4) * ROWPAD;     // scalar j-column base
            v8f acc = {};
#pragma unroll
            for (int c = 0; c < 4; ++c) {
                v2f b;
                b[0] = lx[jbase + ioff + (c * 4    ) * IBLK];
                b[1] = lx[jbase + ioff + (c * 4 + 1) * IBLK];
                acc = __builtin_amdgcn_wmma_f32_16x16x4_f32(
                    false, hbs[c], false, b, (short)0, acc, false, false);
            }
#pragma unroll
            for (int v = 0; v < 8; ++v)
                lx[jbase + doff + v * IBLK] = acc[v];
        }
        __syncthreads();   // next pass's stage K (or the store) reads all rows
    }

    // ---- store (coalesced float4) -----------------------------------------
    float* __restrict__ orow = out + (size_t)token * DIM;
#pragma unroll
    for (int it = 0; it < 8; ++it) {
        const int g = (it * 128 + tid) * 4;
        const int base = (g >> 8) * IBLK + ((g >> 4) & 15) * ROWPAD + (g & 15);
        float4 o;
        o.x = lx[base + 0];  o.y = lx[base + 1];
        o.z = lx[base + 2];  o.w = lx[base + 3];
        *(float4*)(orow + g) = o;
    }
}

extern "C" void kernel_launch(void* const* d_in, const int* in_sizes, int n_in,
                              void* d_out, int out_size, void* d_ws, size_t ws_size,
                              hipStream_t stream) {
    const float* x     = (const float*)d_in[0];
    const float* signs = (const float*)d_in[1];
    float* out         = (float*)d_out;
    const int ntok = in_sizes[0] / DIM;     // 8192 tokens
    fwht2_kernel<<<dim3(ntok), dim3(128), 0, stream>>>(x, signs, out);
}